// Gate_59227599012428
// MI455X (gfx1250) — compile-verified
//
#include <hip/hip_runtime.h>
#include <hip/hip_bf16.h>
#include <math.h>

typedef __attribute__((ext_vector_type(2))) float v2f;
typedef __attribute__((ext_vector_type(8))) float v8f;

#define N_EXPERTS   256
#define N_GROUPS    8
#define GROUP_SZ    32
#define TOPK_GROUPS 4
#define TOPK        8
#define ROUTE_SCALE 2.5f
#define DIM         7168
#define BLOCK_M     64    // tokens per block (4 M-tiles of 16)
#define KC          16    // K chunk (double-buffered)
#define NCHUNK      (DIM / KC)
#define APAD        20    // row stride (words): 80B = 5x16B (b128-aligned), 20*l mod 64 distinct for l=0..15
#define SPAD        257   // odd stride -> conflict-free per-token rows in epilogue

struct Stage {
    float A[BLOCK_M][APAD];     // 64 x 20 x 4B  = 5.0 KB
    float B[N_EXPERTS][APAD];   // 256 x 20 x 4B = 20.0 KB
};
union Smem {
    Stage st[2];                // 50 KB double-buffered GEMM staging
    float scores[32][SPAD];     // 32.1 KB (one 32-token half at a time)
};

// CDNA5 async copy: memory -> LDS, no VGPR data round-trip, tracked by ASYNCcnt.
__device__ __forceinline__ void async_b128(unsigned lds_off, const float* gaddr) {
    asm volatile("global_load_async_to_lds_b128 %0, %1, off"
                 :: "v"(lds_off), "v"(gaddr) : "memory");
}
template <int N>
__device__ __forceinline__ void wait_asynccnt() {
    asm volatile("s_wait_asynccnt %0" :: "i"(N) : "memory");
}

__device__ __forceinline__ void stage_chunk(Stage* st,
                                            const float* __restrict__ x,
                                            const float* __restrict__ weight,
                                            int tokBase, int T, int k0, int tid) {
    // A chunk: 64 rows x 16 K -> one b128 per thread
    {
        int row = tid >> 2;
        int col = (tid & 3) * 4;
        int gr  = tokBase + row; if (gr >= T) gr = T - 1;
        async_b128((unsigned)(size_t)&st->A[row][col],
                   x + (size_t)gr * DIM + k0 + col);
    }
    // B chunk: 256 experts x 16 K -> four b128 per thread
    #pragma unroll
    for (int i = 0; i < 4; ++i) {
        int linear = tid + i * 256;
        int row = linear >> 2;
        int col = (linear & 3) * 4;
        async_b128((unsigned)(size_t)&st->B[row][col],
                   weight + (size_t)row * DIM + k0 + col);
    }
}

__global__ __launch_bounds__(256)
void moe_gate_kernel(const float* __restrict__ x,
                     const float* __restrict__ weight,
                     const float* __restrict__ bias,
                     float* __restrict__ out, int T)
{
    __shared__ Smem sm;
    __shared__ float biasS[N_EXPERTS];

    const int tid  = threadIdx.x;
    const int lane = tid & 31;
    const int wave = tid >> 5;
    const int mw   = wave >> 2;          // 0..1 : rows mw*32 .. mw*32+31 (two 16-row tiles)
    const int nw   = wave & 3;           // 0..3 : cols nw*64 .. nw*64+63 (four 16-col tiles)
    const int l16  = lane & 15;
    const int koff = (lane >> 4) * 2;    // lanes 16-31 hold K+2,K+3
    const int tokBase = blockIdx.x * BLOCK_M;

    if (tid < N_EXPERTS) biasS[tid] = bias[tid];

    const v8f vzero = {0.f,0.f,0.f,0.f,0.f,0.f,0.f,0.f};
    v8f acc[2][4];
    #pragma unroll
    for (int i = 0; i < 2; ++i)
        #pragma unroll
        for (int t = 0; t < 4; ++t) acc[i][t] = vzero;

    // ---------------- GEMM: scores = x_tile @ W^T (double-buffered async) ----------------
    stage_chunk(&sm.st[0], x, weight, tokBase, T, 0, tid);   // prologue: chunk 0 in flight

    int buf = 0;
    for (int chunk = 0; chunk < NCHUNK; ++chunk) {
        const int nbuf = buf ^ 1;
        __syncthreads();   // all waves done READING stage[nbuf] (chunk-1's compute)

        if (chunk + 1 < NCHUNK) {
            stage_chunk(&sm.st[nbuf], x, weight, tokBase, T, (chunk + 1) * KC, tid);
            wait_asynccnt<5>();   // drain my 5 older ops (chunk i); leave prefetch in flight
        } else {
            wait_asynccnt<0>();
        }
        __syncthreads();   // everyone's chunk-i LDS writes visible

        const Stage* st = &sm.st[buf];
        #pragma unroll
        for (int ks = 0; ks < KC; ks += 4) {
            v2f a0 = *(const v2f*)&st->A[mw * 32 +      l16][ks + koff];
            v2f a1 = *(const v2f*)&st->A[mw * 32 + 16 + l16][ks + koff];
            #pragma unroll
            for (int t = 0; t < 4; ++t) {
                v2f b = *(const v2f*)&st->B[nw * 64 + t * 16 + l16][ks + koff];
                acc[0][t] = __builtin_amdgcn_wmma_f32_16x16x4_f32(
                    false, a0, false, b, (short)0, acc[0][t], false, false);
                acc[1][t] = __builtin_amdgcn_wmma_f32_16x16x4_f32(
                    false, a1, false, b, (short)0, acc[1][t], false, false);
            }
        }
        buf = nbuf;
    }

    // ---------------- epilogue: two 32-token halves ----------------
    // C/D layout: lanes 0-15 -> M = v, lanes 16-31 -> M = 8+v; N = lane%16
    #pragma unroll
    for (int h = 0; h < 2; ++h) {
        __syncthreads();   // GEMM LDS reads done / previous half's routing done

        if (mw == h) {
            #pragma unroll
            for (int i = 0; i < 2; ++i) {
                #pragma unroll
                for (int t = 0; t < 4; ++t) {
                    #pragma unroll
                    for (int v = 0; v < 8; ++v) {
                        int row = i * 16 + (lane >> 4) * 8 + v;   // 0..31 within half
                        int col = nw * 64 + t * 16 + l16;
                        float sc = acc[i][t][v];
                        sm.scores[row][col] = 1.0f / (1.0f + __expf(-sc));
                    }
                }
            }
        }
        __syncthreads();

        // routing: one token per thread, threads 0..31
        if (tid < 32) {
            int tok = tokBase + h * 32 + tid;
            if (tok < T) {
                // group score = sum of top-2 bias-corrected scores per group of 32
                float gsc[N_GROUPS];
                #pragma unroll
                for (int g = 0; g < N_GROUPS; ++g) {
                    float m1 = -INFINITY, m2 = -INFINITY;
                    for (int j = 0; j < GROUP_SZ; ++j) {
                        float v = sm.scores[tid][g * GROUP_SZ + j] + biasS[g * GROUP_SZ + j];
                        if (v > m1) { m2 = m1; m1 = v; }
                        else if (v > m2) { m2 = v; }
                    }
                    gsc[g] = m1 + m2;
                }
                // top-4 groups (ties -> lowest index, matching lax.top_k)
                unsigned keepMask = 0;
                #pragma unroll
                for (int r = 0; r < TOPK_GROUPS; ++r) {
                    float best = -INFINITY; int bi = 0;
                    for (int g = 0; g < N_GROUPS; ++g)
                        if (!((keepMask >> g) & 1u) && gsc[g] > best) { best = gsc[g]; bi = g; }
                    keepMask |= 1u << bi;
                }
                // top-8 experts over kept groups, (value, -index) lexicographic
                float pv = INFINITY; int pi = -1;
                float w[TOPK]; int idx[TOPK];
                float wsum = 0.0f;
                for (int r = 0; r < TOPK; ++r) {
                    float bv = -INFINITY; int bi = 0;
                    for (int e = 0; e < N_EXPERTS; ++e) {
                        if (!((keepMask >> (e >> 5)) & 1u)) continue;
                        float v = sm.scores[tid][e] + biasS[e];
                        if ((v < pv || (v == pv && e > pi)) && v > bv) { bv = v; bi = e; }
                    }
                    idx[r] = bi; pv = bv; pi = bi;
                    float ow = sm.scores[tid][bi];   // weight from ORIGINAL sigmoid score
                    w[r] = ow; wsum += ow;
                }
                float inv = ROUTE_SCALE / wsum;
                float* outw = out + (size_t)tok * TOPK;
                float* outi = out + (size_t)T * TOPK + (size_t)tok * TOPK;
                #pragma unroll
                for (int r = 0; r < TOPK; ++r) {
                    outw[r] = w[r] * inv;
                    outi[r] = (float)idx[r];
                }
            }
        }
    }
}

extern "C" void kernel_launch(void* const* d_in, const int* in_sizes, int n_in,
                              void* d_out, int out_size, void* d_ws, size_t ws_size,
                              hipStream_t stream) {
    const float* x      = (const float*)d_in[0];
    const float* weight = (const float*)d_in[1];
    const float* bias   = (const float*)d_in[2];
    float* out = (float*)d_out;
    int T = in_sizes[0] / DIM;
    int grid = (T + BLOCK_M - 1) / BLOCK_M;
    moe_gate_kernel<<<grid, 256, 0, stream>>>(x, weight, bias, out, T);
}